// MAMDense_36137854828779
// MI455X (gfx1250) — compile-verified
//
#include <hip/hip_runtime.h>

// MAM dense: C[i,j] = max_k(A[i,k]*W[j,k]) + min_k(A[i,k]*W[j,k]) + bias[j]
// A = x flattened [M,K], W = weight [N,K] (row-major, k contiguous), C = [M,N].
// Max/min-of-products semiring => no WMMA path exists; VALU-bound kernel.
// Data movement: gfx1250 GLOBAL_LOAD_ASYNC_TO_LDS_B128 double-buffer pipeline
// (ASYNCcnt-tracked), with a register-staged fallback if the builtin is absent.

#define BM 64
#define BN 64
#define KC 32
#define LDK 36   // padded LDS row stride (floats): 16B-aligned, bank-conflict-free reads
#define NT 256

#if defined(__HIP_DEVICE_COMPILE__) && __has_builtin(__builtin_amdgcn_global_load_async_to_lds_b128)
#define USE_ASYNC_LDS 1
#else
#define USE_ASYNC_LDS 0
#endif

#if USE_ASYNC_LDS
// Builtin prototype (from clang diagnostic): params are pointers to <4 x i32>,
// src in AS1 (printed '__device__'), dst in AS3 ('__shared__').
typedef int v4i __attribute__((ext_vector_type(4)));
typedef __attribute__((address_space(1))) v4i gv4i;
typedef __attribute__((address_space(3))) v4i lv4i;

// generic->AS1: numerically identity. generic->AS3: LDS offset is addr[31:0]
// per the aperture rules, so truncation through an integer is exact.
static __device__ __forceinline__ void async_b128(const void* g, void* l) {
    __builtin_amdgcn_global_load_async_to_lds_b128(
        (gv4i*)(unsigned long long)g,
        (lv4i*)(unsigned int)(unsigned long long)l,
        /*offset=*/0, /*cpol=*/0);
}
static __device__ __forceinline__ void async_wait0() {
#if __has_builtin(__builtin_amdgcn_s_wait_asynccnt)
    __builtin_amdgcn_s_wait_asynccnt(0);
#else
    asm volatile("s_wait_asynccnt 0x0" ::: "memory");
#endif
}
#endif

// min-2-blocks/WGP hint: caps register budget under 256 VGPRs, removing
// s_set_vgpr_msb traffic from the inner loop and securing occupancy.
__global__ __launch_bounds__(NT, 2)
void mam_dense_kernel(const float* __restrict__ A,    // [M,K]
                      const float* __restrict__ W,    // [N,K]
                      const float* __restrict__ bias, // [N]
                      float* __restrict__ C,          // [M,N]
                      int M, int N, int K)
{
    __shared__ float As[2][BM * LDK];
    __shared__ float Ws[2][BN * LDK];

    const int tid = threadIdx.x;
    const int tx  = tid & 15;   // column group (cyclic cols: bj + tx + 16*c)
    const int ty  = tid >> 4;   // row group    (blocked rows: bi + ty*4 + r)
    const int bi  = blockIdx.y * BM;
    const int bj  = blockIdx.x * BN;

    // Global->LDS copy mapping: 8 threads per row, one b128 each,
    // two rows per thread (lrow and lrow+32). 8 lanes = 128B contiguous.
    const int lrow = tid >> 3;         // 0..31
    const int lcol = (tid & 7) << 2;   // 0,4,...,28

    const float*  Ap    = A + (size_t)(bi + lrow) * K + lcol;
    const float*  Wp    = W + (size_t)(bj + lrow) * K + lcol;
    const size_t  rstep = (size_t)32 * K;

    const int ldsOffLo = lrow * LDK + lcol;
    const int ldsOffHi = (lrow + 32) * LDK + lcol;

    float vmax[16], vmin[16];
#pragma unroll
    for (int i = 0; i < 16; ++i) {
        vmax[i] = -__builtin_inff();
        vmin[i] =  __builtin_inff();
    }

#if USE_ASYNC_LDS
    // Prologue: async-copy chunk 0 straight into LDS buffer 0.
    async_b128(Ap,         &As[0][ldsOffLo]);
    async_b128(Ap + rstep, &As[0][ldsOffHi]);
    async_b128(Wp,         &Ws[0][ldsOffLo]);
    async_b128(Wp + rstep, &Ws[0][ldsOffHi]);
    async_wait0();
#else
    float4 a0 = *(const float4*)(Ap);
    float4 a1 = *(const float4*)(Ap + rstep);
    float4 w0 = *(const float4*)(Wp);
    float4 w1 = *(const float4*)(Wp + rstep);
    *(float4*)&As[0][ldsOffLo] = a0;
    *(float4*)&As[0][ldsOffHi] = a1;
    *(float4*)&Ws[0][ldsOffLo] = w0;
    *(float4*)&Ws[0][ldsOffHi] = w1;
#endif
    __syncthreads();

    const int nchunk = K / KC;   // 768/32 = 24
    int cur = 0;
    for (int t = 0; t < nchunk; ++t) {
        const int  nxt  = cur ^ 1;
        const bool more = (t + 1) < nchunk;

        // Kick off chunk t+1 into the other buffer; compute below hides it.
        if (more) {
            const float* ap = Ap + (size_t)(t + 1) * KC;
            const float* wp = Wp + (size_t)(t + 1) * KC;
#if USE_ASYNC_LDS
            async_b128(ap,         &As[nxt][ldsOffLo]);
            async_b128(ap + rstep, &As[nxt][ldsOffHi]);
            async_b128(wp,         &Ws[nxt][ldsOffLo]);
            async_b128(wp + rstep, &Ws[nxt][ldsOffHi]);
#else
            a0 = *(const float4*)(ap);
            a1 = *(const float4*)(ap + rstep);
            w0 = *(const float4*)(wp);
            w1 = *(const float4*)(wp + rstep);
#endif
            if (t + 2 < nchunk) {
                __builtin_prefetch(Ap + (size_t)(t + 2) * KC, 0, 0); // global_prefetch_b8
                __builtin_prefetch(Wp + (size_t)(t + 2) * KC, 0, 0);
            }
        }

        // Compute on current LDS buffer: 4 k's per step via b128 loads.
#pragma unroll
        for (int kk = 0; kk < KC; kk += 4) {
            float4 av[4], wv[4];
#pragma unroll
            for (int r = 0; r < 4; ++r)
                av[r] = *(const float4*)&As[cur][(ty * 4 + r) * LDK + kk];   // broadcast
#pragma unroll
            for (int c = 0; c < 4; ++c)
                wv[c] = *(const float4*)&Ws[cur][(tx + 16 * c) * LDK + kk];  // conflict-free
#pragma unroll
            for (int r = 0; r < 4; ++r) {
#pragma unroll
                for (int c = 0; c < 4; ++c) {
                    const float p0 = av[r].x * wv[c].x;
                    const float p1 = av[r].y * wv[c].y;
                    const float p2 = av[r].z * wv[c].z;
                    const float p3 = av[r].w * wv[c].w;
                    const int idx = r * 4 + c;
                    vmax[idx] = fmaxf(vmax[idx], fmaxf(fmaxf(p0, p1), fmaxf(p2, p3)));
                    vmin[idx] = fminf(vmin[idx], fminf(fminf(p0, p1), fminf(p2, p3)));
                }
            }
        }

        if (more) {
#if USE_ASYNC_LDS
            async_wait0();       // this wave's chunk t+1 copies have landed
#else
            *(float4*)&As[nxt][ldsOffLo] = a0;
            *(float4*)&As[nxt][ldsOffHi] = a1;
            *(float4*)&Ws[nxt][ldsOffLo] = w0;
            *(float4*)&Ws[nxt][ldsOffHi] = w1;
#endif
            __syncthreads();     // everyone's copies landed; all reads of cur done
        }
        cur = nxt;
    }

    // Epilogue: bias add + store (half-wave-contiguous 64B stores per (r,c)).
    float bv[4];
#pragma unroll
    for (int c = 0; c < 4; ++c) bv[c] = bias[bj + tx + 16 * c];
#pragma unroll
    for (int r = 0; r < 4; ++r) {
        float* crow = C + (size_t)(bi + ty * 4 + r) * N + bj;
#pragma unroll
        for (int c = 0; c < 4; ++c)
            crow[tx + 16 * c] = vmax[r * 4 + c] + vmin[r * 4 + c] + bv[c];
    }
}

extern "C" void kernel_launch(void* const* d_in, const int* in_sizes, int n_in,
                              void* d_out, int out_size, void* d_ws, size_t ws_size,
                              hipStream_t stream) {
    (void)n_in; (void)out_size; (void)d_ws; (void)ws_size;
    const float* x    = (const float*)d_in[0];  // [2,1024,768] -> [M,K]
    const float* w    = (const float*)d_in[1];  // [N,K]
    const float* bias = (const float*)d_in[2];  // [N]
    float*       out  = (float*)d_out;          // [M,N]

    const int N = in_sizes[2];            // 768
    const int K = in_sizes[1] / N;        // 768
    const int M = in_sizes[0] / K;        // 2048

    dim3 grid(N / BN, M / BM);            // (12, 32)
    mam_dense_kernel<<<grid, dim3(NT), 0, stream>>>(x, w, bias, out, M, N, K);
}